// ASIC_24386824307058
// MI455X (gfx1250) — compile-verified
//
#include <hip/hip_runtime.h>
#include <hip/hip_bf16.h>

// ---------------------------------------------------------------------------
// CDNA5 (gfx1250) implementation of the 4-layer differentiable-logic grid.
//
// Key algebra: for one pixel,   result = sum_c tw_c * prod_n phi_{bit}(v_n)
// with phi_0(v)=1-v, phi_1(v)=v.  Rewriting in the monomial basis:
//     result = sum_{S subset of 4 neighbors} coef_S * prod_{n in S} v_n
//     coef   = M @ sigmoid(toggle_gates[layer])        (M: fixed 16x16, {-1,0,1})
// The basis change is a genuine (16x16)x(16x65536) GEMM per layer -> done with
// V_WMMA_F32_16X16X4_F32 (f32 in/out, exact for M's +-1 entries).  The eval is
// then 11 mul + 15 fma per (pixel,batch) lane against wave-uniform coefs.
//
// Data layout: state kept as (H, W, B=32) floats, batch innermost, so one
// wave32 = one pixel, lane = batch element, and every neighbor load is one
// coalesced 128B global_load_b32.  Whole working set (~38MB) lives in L2.
// ---------------------------------------------------------------------------

typedef float v2f __attribute__((ext_vector_type(2)));
typedef float v8f __attribute__((ext_vector_type(8)));

#define H 256
#define W 256
#define P 65536          // H*W
#define NC 16            // combos
#define B 32             // batch
#define XH 128
#define XW 128

// Mobius matrix entry: M[S,c] = (c subset of S) ? (-1)^popc(S^c) : 0
// Bit convention: bit (3-n) of the 4-bit index corresponds to neighbor n,
// matching itertools.product MSB-first enumeration in the reference.
__device__ __forceinline__ float mval(int S, int c) {
    return (c & ~S & 15) ? 0.0f : ((__popc(S ^ c) & 1) ? -1.0f : 1.0f);
}

// sigmoid via hardware transcendentals: v_exp_f32 + v_rcp_f32 (~1 ulp each),
// avoiding the ~10-op IEEE div_scale/div_fmas/div_fixup expansion.
__device__ __forceinline__ float sigm(float t) {
    return __builtin_amdgcn_rcpf(1.0f + __expf(-t));
}

// ---------------------------------------------------------------------------
// Transform: coef[p][S] = sum_c M[S,c] * sigmoid(tg[c][p]) for one layer.
// One wave handles a 16-pixel tile with 4 chained V_WMMA_F32_16X16X4_F32.
//   A (16x4 f32, documented layout): lane l -> row M=l&15, VGPR v -> K=v+2*(l>>4)
//   B (4x16  f32, symmetric layout): lane l -> col N=l&15, VGPR v -> K=v+2*(l>>4)
//   D (16x16 f32, documented layout): VGPR r, lane l -> row M=r+8*(l>>4), col N=l&15
// M dimension = monomial index S, N dimension = pixel within tile, K = combo c.
// ---------------------------------------------------------------------------
__global__ void __launch_bounds__(256)
logic_transform(const float* __restrict__ tg, float* __restrict__ coef) {
    const int wave = blockIdx.x * (blockDim.x >> 5) + (threadIdx.x >> 5);
    const int lane = threadIdx.x & 31;
    const int tileBase = wave << 4;            // 16 consecutive pixels
    const int n  = lane & 15;                  // pixel col (B/D) == row S (A)
    const int kh = lane >> 4;                  // K half select

    v8f d = {0.f, 0.f, 0.f, 0.f, 0.f, 0.f, 0.f, 0.f};

#pragma unroll
    for (int k = 0; k < 4; ++k) {
        const int c0 = 4 * k + 2 * kh;         // combo index for VGPR0 slot
        // B operand: sigmoid(gates) for combos c0,c0+1 at 16 tile pixels
        float b0 = sigm(tg[c0 * P + tileBase + n]);
        float b1 = sigm(tg[(c0 + 1) * P + tileBase + n]);
        v2f Bm = {b0, b1};
        // A operand: Mobius matrix columns c0, c0+1 (rows = S = n)
        v2f Am = {mval(n, c0), mval(n, c0 + 1)};
        d = __builtin_amdgcn_wmma_f32_16x16x4_f32(
            /*neg_a=*/false, Am, /*neg_b=*/false, Bm,
            /*c_mod=*/(short)0, d, /*reuse_a=*/false, /*reuse_b=*/false);
    }

    // Scatter D to coef[pixel][S]: lane holds S = kh*8 + r for r=0..7, contiguous.
    float* dst = coef + ((tileBase + n) << 4) + (kh << 3);
    *(float4*)(dst)     = make_float4(d[0], d[1], d[2], d[3]);
    *(float4*)(dst + 4) = make_float4(d[4], d[5], d[6], d[7]);
}

// Shared multilinear evaluation: 11 mul + 15 fma + clamp.
__device__ __forceinline__ float eval_poly(const float* __restrict__ cp,
                                           float v0, float v1, float v2, float v3) {
    float4 ca = *(const float4*)(cp);
    float4 cb = *(const float4*)(cp + 4);
    float4 cc = *(const float4*)(cp + 8);
    float4 cd = *(const float4*)(cp + 12);
    float t23 = v2 * v3, t01 = v0 * v1;
    float r = ca.x;
    r = fmaf(v3,        ca.y, r);
    r = fmaf(v2,        ca.z, r);
    r = fmaf(t23,       ca.w, r);
    r = fmaf(v1,        cb.x, r);
    r = fmaf(v1 * v3,   cb.y, r);
    r = fmaf(v1 * v2,   cb.z, r);
    r = fmaf(v1 * t23,  cb.w, r);
    r = fmaf(v0,        cc.x, r);
    r = fmaf(v0 * v3,   cc.y, r);
    r = fmaf(v0 * v2,   cc.z, r);
    r = fmaf(v0 * t23,  cc.w, r);
    r = fmaf(t01,       cd.x, r);
    r = fmaf(t01 * v3,  cd.y, r);
    r = fmaf(t01 * v2,  cd.z, r);
    r = fmaf(t01 * t23, cd.w, r);
    return fminf(fmaxf(r, 0.0f), 1.0f);
}

// Layer 0: state0 is x scattered on (even,even) pixels -> exactly one of the
// four 2x2 neighbors is nonzero: result = clamp(coef[0] + coef[8>>n*] * x).
__global__ void __launch_bounds__(256)
logic_eval_first(const float* __restrict__ x, const float* __restrict__ coef,
                 float* __restrict__ out) {
    const int p    = blockIdx.x * 8 + (threadIdx.x >> 5);
    const int lane = threadIdx.x & 31;         // batch element
    const int h = p >> 8, w = p & 255;
    const int nstar = ((h & 1) << 1) | (w & 1);
    const int hi = ((h + (h & 1)) >> 1) & 127;
    const int wi = ((w + (w & 1)) >> 1) & 127;
    const float v = x[lane * (XH * XW) + hi * XW + wi];
    const float* cp = coef + (p << 4);
    const float r = fminf(fmaxf(fmaf(v, cp[8 >> nstar], cp[0]), 0.0f), 1.0f);
    out[(p << 5) + lane] = r;
}

// Middle layers: full 256x256 grid, wrap-around neighbors.
__global__ void __launch_bounds__(256)
logic_eval_mid(const float* __restrict__ st, const float* __restrict__ coef,
               float* __restrict__ out) {
    const int p    = blockIdx.x * 8 + (threadIdx.x >> 5);
    const int lane = threadIdx.x & 31;
    const int h = p >> 8, w = p & 255;
    const int h1 = (h + 1) & 255, w1 = (w + 1) & 255;
    const float v0 = st[(((h  << 8) | w ) << 5) + lane];
    const float v1 = st[(((h  << 8) | w1) << 5) + lane];
    const float v2 = st[(((h1 << 8) | w ) << 5) + lane];
    const float v3 = st[(((h1 << 8) | w1) << 5) + lane];
    out[(p << 5) + lane] = eval_poly(coef + (p << 4), v0, v1, v2, v3);
}

// Last layer: only (even,even) pixels survive the final ::2 slice, so compute
// just those 128x128 and write straight to d_out in (B,128,128) layout.
__global__ void __launch_bounds__(256)
logic_eval_last(const float* __restrict__ st, const float* __restrict__ coef,
                float* __restrict__ out) {
    const int q    = blockIdx.x * 8 + (threadIdx.x >> 5);   // 0..16383
    const int lane = threadIdx.x & 31;
    const int i = q >> 7, j = q & 127;
    const int p = ((i << 1) << 8) | (j << 1);               // h=2i, w=2j (no wrap needed)
    const float v0 = st[((p      ) << 5) + lane];
    const float v1 = st[((p +   1) << 5) + lane];
    const float v2 = st[((p + 256) << 5) + lane];
    const float v3 = st[((p + 257) << 5) + lane];
    out[lane * (XH * XW) + q] = eval_poly(coef + (p << 4), v0, v1, v2, v3);
}

extern "C" void kernel_launch(void* const* d_in, const int* in_sizes, int n_in,
                              void* d_out, int out_size, void* d_ws, size_t ws_size,
                              hipStream_t stream) {
    const float* x  = (const float*)d_in[0];   // (32,128,128)
    const float* tg = (const float*)d_in[1];   // (4,16,256,256)
    float* out = (float*)d_out;                // (32,128,128)

    char* ws = (char*)d_ws;
    float* coef = (float*)ws;                          //  4 MiB: 65536 x 16
    float* sA   = (float*)(ws + (4u << 20));           //  8 MiB: 65536 x 32
    float* sB   = (float*)(ws + (12u << 20));          //  8 MiB: 65536 x 32

    const dim3 blk(256);
    const int TGRID = 512;     // 4096 tiles / 8 waves per block
    const int EGRID = 8192;    // 65536 pixels / 8 waves per block
    const int LGRID = 2048;    // 16384 pixels / 8 waves per block
    const long LSTRIDE = (long)NC * P;   // gates per layer

    logic_transform<<<TGRID, blk, 0, stream>>>(tg + 0 * LSTRIDE, coef);
    logic_eval_first<<<EGRID, blk, 0, stream>>>(x, coef, sA);

    logic_transform<<<TGRID, blk, 0, stream>>>(tg + 1 * LSTRIDE, coef);
    logic_eval_mid<<<EGRID, blk, 0, stream>>>(sA, coef, sB);

    logic_transform<<<TGRID, blk, 0, stream>>>(tg + 2 * LSTRIDE, coef);
    logic_eval_mid<<<EGRID, blk, 0, stream>>>(sB, coef, sA);

    logic_transform<<<TGRID, blk, 0, stream>>>(tg + 3 * LSTRIDE, coef);
    logic_eval_last<<<LGRID, blk, 0, stream>>>(sA, coef, out);
}